// RectUnsupFocalLoss_60438779789554
// MI455X (gfx1250) — compile-verified
//
#include <hip/hip_runtime.h>
#include <hip/hip_bf16.h>

// Problem constants (from reference): logits [N=8, C=19, H=512, W=1024], fp32.
#define N_   8
#define C_   19
#define H_   512
#define W_   1024
#define HW_  (H_ * W_)            // 524288 = 2^19
#define CHW_ (C_ * HW_)
#define P_   (N_ * HW_)           // 4194304 pixels (mean denominator)

#define TPB      256
#define NBLK     2048
#define NTHREADS (TPB * NBLK)     // 524288 -> 8 pixels per thread, uniform

typedef float v2f __attribute__((ext_vector_type(2)));
typedef float v8f __attribute__((ext_vector_type(8)));

// Wave32 all-lanes reduction using V_WMMA_F32_16X16X4_F32 (f32-exact).
// A = ones(16x4), B = {x, 0} (4x16): D[m][n] = sum_k B[k][n] = x[n] + x[n+16]
// for every m. c[0] in lane l therefore holds colsum(l%16), replicated in both
// half-waves; 4 xor-shuffles over the 16 distinct values finish the sum.
// Requires EXEC == all ones (uniform control flow at call sites).
__device__ __forceinline__ float wave_reduce_wmma(float x) {
  v2f a; a.x = 1.0f; a.y = 1.0f;   // A-matrix: all ones
  v2f b; b.x = x;    b.y = 0.0f;   // B-matrix rows k=0,1 = lane values; k=2,3 = 0
  v8f c = {};
  c = __builtin_amdgcn_wmma_f32_16x16x4_f32(
      /*neg_a=*/false, a, /*neg_b=*/false, b,
      /*c_mod=*/(short)0, c, /*reuse_a=*/false, /*reuse_b=*/false);
  float s = c[0];
  s += __shfl_xor(s, 1, 32);
  s += __shfl_xor(s, 2, 32);
  s += __shfl_xor(s, 4, 32);
  s += __shfl_xor(s, 8, 32);
  return s;  // every lane: sum of all 32 inputs
}

__global__ __launch_bounds__(TPB) void focal_loss_partial_kernel(
    const float* __restrict__ LB, const float* __restrict__ LA,
    float* __restrict__ partial) {
  const int tid = blockIdx.x * TPB + threadIdx.x;
  float acc = 0.0f;

  // P_ % NTHREADS == 0 -> every thread runs exactly 8 iterations (uniform EXEC).
  for (int p = tid; p < P_; p += NTHREADS) {
    const int n  = p >> 19;          // p / HW_
    const int hw = p & (HW_ - 1);    // p % HW_
    const float* bptr = LB + (size_t)n * CHW_ + hw;
    const float* aptr = LA + (size_t)n * CHW_ + hw;

    // Stream the 19-class column; non-temporal (no reuse, 637MB >> 192MB L2).
    float xb[C_], xa[C_];
#pragma unroll
    for (int c = 0; c < C_; ++c) {
      xb[c] = __builtin_nontemporal_load(bptr + (size_t)c * HW_);
      xa[c] = __builtin_nontemporal_load(aptr + (size_t)c * HW_);
    }

    // max + first-occurrence argmax (teacher), max (student)
    float mb = xb[0]; int idx = 0;
    float ma = xa[0];
#pragma unroll
    for (int c = 1; c < C_; ++c) {
      if (xb[c] > mb) { mb = xb[c]; idx = c; }
      ma = fmaxf(ma, xa[c]);
    }

    float sb = 0.0f, sa = 0.0f, acc1 = 0.0f, acc2 = 0.0f;
    float ebi = 0.0f, eai = 0.0f;
#pragma unroll
    for (int c = 0; c < C_; ++c) {
      const float tb = xb[c] - mb;
      const float ta = xa[c] - ma;
      const float eb = __expf(tb);
      const float ea = __expf(ta);
      sb += eb;
      sa += ea;
      acc1 = fmaf(eb, tb, acc1);     // -> sum pb*log pb (un-normalized part)
      acc2 = fmaf(eb, ta, acc2);     // -> sum pb*log pa (un-normalized part)
      const bool sel = (c == idx);
      ebi = sel ? eb : ebi;
      eai = sel ? ea : eai;
    }

    const float rsb = 1.0f / sb;
    const float rsa = 1.0f / sa;
    const float S1  = acc1 * rsb - __logf(sb);   // sum_c pb*log(pb)
    const float S2  = acc2 * rsb - __logf(sa);   // sum_c pb*log(pa)
    const float focal = fabsf(ebi * rsb - eai * rsa);
    // (kldiv + entropy) sums telescope: S1 - focal*S2 - S1 = -focal*S2
    float loss = -focal * S2;
    loss *= __expf(S1);                          // * exp(-entropy_detached)
    acc += fmaxf(loss, 1e-8f);                   // clip then accumulate
  }

  // Deterministic block reduction: WMMA per-wave, then LDS across 8 waves.
  const float wsum = wave_reduce_wmma(acc);
  __shared__ float wpart[TPB / 32];
  const int lane = threadIdx.x & 31;
  const int wid  = threadIdx.x >> 5;
  if (lane == 0) wpart[wid] = wsum;
  __syncthreads();
  if (threadIdx.x == 0) {
    float t = 0.0f;
#pragma unroll
    for (int i = 0; i < TPB / 32; ++i) t += wpart[i];
    partial[blockIdx.x] = t;
  }
}

__global__ void final_reduce_kernel(const float* __restrict__ partial,
                                    float* __restrict__ out) {
  // Single wave of 32 lanes; fixed-order serial sums then WMMA reduction.
  float s = 0.0f;
  for (int i = threadIdx.x; i < NBLK; i += 32) s += partial[i];
  const float tot = wave_reduce_wmma(s);
  if (threadIdx.x == 0) out[0] = tot * (1.0f / (float)P_);  // exact *2^-22
}

extern "C" void kernel_launch(void* const* d_in, const int* in_sizes, int n_in,
                              void* d_out, int out_size, void* d_ws, size_t ws_size,
                              hipStream_t stream) {
  (void)in_sizes; (void)n_in; (void)out_size; (void)ws_size;
  const float* logits_before = (const float*)d_in[0];
  const float* logits_after  = (const float*)d_in[1];
  float* partial = (float*)d_ws;          // NBLK floats = 8 KB of scratch
  float* out     = (float*)d_out;         // single fp32 scalar

  focal_loss_partial_kernel<<<NBLK, TPB, 0, stream>>>(logits_before,
                                                      logits_after, partial);
  final_reduce_kernel<<<1, 32, 0, stream>>>(partial, out);
}